// SetAbstraction_38585986187947
// MI455X (gfx1250) — compile-verified
//
#include <hip/hip_runtime.h>

typedef float v2f __attribute__((ext_vector_type(2)));
typedef float v8f __attribute__((ext_vector_type(8)));

#define NB 8
#define NPTS 16384
#define NPOINT 1024
#define NSAMP 32
#define MROWS (NB * NPOINT * NSAMP)   // 262144 rows of the MLP "GEMM"
#define BPC (NB * NPOINT)             // 8192 centroids
#define RAD2 0.04f                    // 0.2^2
#define BN_EPS 1e-5f

// ---- workspace layout (in floats) ----
// A0   : MROWS*16   (layer-0 input, K padded 9->16)
// Y0   : MROWS*64   (layer-0 raw pre-BN output)
// Y1   : MROWS*64   (layer-1 raw pre-BN output)
// GMAX : BPC*128    (per (b,p,ch) max over 32 samples of raw layer-2 out)
// GMIN : BPC*128
// STATS: 3 layers * 256 (sum[128], sumsq[128] slots each)
// AB   : 3 layers * 128 float2 (per-channel fused BN affine a,c)
static constexpr size_t OFF_A0    = 0;
static constexpr size_t OFF_Y0    = OFF_A0 + (size_t)MROWS * 16;
static constexpr size_t OFF_Y1    = OFF_Y0 + (size_t)MROWS * 64;
static constexpr size_t OFF_GMAX  = OFF_Y1 + (size_t)MROWS * 64;
static constexpr size_t OFF_GMIN  = OFF_GMAX + (size_t)BPC * 128;
static constexpr size_t OFF_STATS = OFF_GMIN + (size_t)BPC * 128;
static constexpr size_t OFF_AB    = OFF_STATS + 3 * 256;

__device__ __forceinline__ float relu_f(float x) { return fmaxf(x, 0.f); }

// ---------------- init: zero the stats accumulators ----------------
__global__ void init_stats_kernel(float* __restrict__ stats) {
  const int i = blockIdx.x * 256 + threadIdx.x;
  if (i < 3 * 256) stats[i] = 0.f;
}

// ---------------- farthest point sampling ----------------
// One 1024-thread block per batch. Each thread owns 16 points in registers.
__global__ __launch_bounds__(1024) void fps_kernel(const float* __restrict__ xyz,
                                                   float* __restrict__ cent) {
  const int b = blockIdx.x;
  const int t = threadIdx.x;
  const float* X = xyz + (size_t)b * NPTS * 3;
  float px[16], py[16], pz[16], dst[16];
#pragma unroll
  for (int j = 0; j < 16; ++j) {
    const int p = t + j * 1024;
    px[j] = X[p * 3 + 0]; py[j] = X[p * 3 + 1]; pz[j] = X[p * 3 + 2];
    dst[j] = 1e10f;
  }
  __shared__ float s_val[32];
  __shared__ int   s_idx[32];
  __shared__ int   s_cur;
  float cx = X[0], cy = X[1], cz = X[2];   // deterministic start index 0
  const int lane = t & 31, w = t >> 5;
  for (int it = 0; it < NPOINT; ++it) {
    if (t == 0) {
      float* o = cent + ((size_t)b * NPOINT + it) * 3;
      o[0] = cx; o[1] = cy; o[2] = cz;
    }
    float bv = -1.f; int bi = NPTS;
#pragma unroll
    for (int j = 0; j < 16; ++j) {
      const float dx = px[j] - cx, dy = py[j] - cy, dz = pz[j] - cz;
      const float nd = fminf(dst[j], dx * dx + dy * dy + dz * dz);
      dst[j] = nd;
      const int p = t + j * 1024;
      if (nd > bv || (nd == bv && p < bi)) { bv = nd; bi = p; }
    }
#pragma unroll
    for (int off = 16; off >= 1; off >>= 1) {  // wave32 argmax, first-index ties
      const float ov = __shfl_xor(bv, off);
      const int   oi = __shfl_xor(bi, off);
      if (ov > bv || (ov == bv && oi < bi)) { bv = ov; bi = oi; }
    }
    if (lane == 0) { s_val[w] = bv; s_idx[w] = bi; }
    __syncthreads();
    if (w == 0) {
      bv = s_val[lane]; bi = s_idx[lane];
#pragma unroll
      for (int off = 16; off >= 1; off >>= 1) {
        const float ov = __shfl_xor(bv, off);
        const int   oi = __shfl_xor(bi, off);
        if (ov > bv || (ov == bv && oi < bi)) { bv = ov; bi = oi; }
      }
      if (lane == 0) s_cur = bi;
    }
    __syncthreads();
    const int cur = s_cur;  // broadcast new farthest; fetch its coords (L0 hit)
    cx = X[cur * 3 + 0]; cy = X[cur * 3 + 1]; cz = X[cur * 3 + 2];
  }
}

// ---------------- ball query + gather + build padded A0 (MROWS x 16) --------
// One wave per centroid; ascending first-32 selection via ballot prefix-sum.
__global__ __launch_bounds__(256) void group_kernel(const float* __restrict__ xyz,
                                                    const float* __restrict__ pts,
                                                    const float* __restrict__ cent,
                                                    float* __restrict__ A0) {
  const int tid = threadIdx.x;
  const int w = tid >> 5, lane = tid & 31;
  const int g = blockIdx.x * 8 + w;           // centroid id 0..8191
  const int b = g >> 10;
  __shared__ int sel[8][NSAMP];
  const float cx = cent[(size_t)g * 3 + 0];
  const float cy = cent[(size_t)g * 3 + 1];
  const float cz = cent[(size_t)g * 3 + 2];
  const float* X = xyz + (size_t)b * NPTS * 3;
  int count = 0;
  for (int chunk = 0; chunk < NPTS / 32; ++chunk) {
    if (count >= NSAMP) break;
    const int j = chunk * 32 + lane;
    const float dx = X[j * 3 + 0] - cx;
    const float dy = X[j * 3 + 1] - cy;
    const float dz = X[j * 3 + 2] - cz;
    const bool pred = (dx * dx + dy * dy + dz * dz) < RAD2;
    const unsigned mask = (unsigned)__ballot(pred);   // wave32: low 32 bits
    const int slot = count + __popc(mask & ((1u << lane) - 1u));
    if (pred && slot < NSAMP) sel[w][slot] = j;
    count += __popc(mask);
  }
  const int filled = count < NSAMP ? count : NSAMP;
  if (lane >= filled) sel[w][lane] = 0;       // pad with index 0 (reference semantics)
  __syncthreads();
  const int idx = sel[w][lane];               // lane == sample slot
  float* o = A0 + ((size_t)g * NSAMP + lane) * 16;
  o[0] = X[idx * 3 + 0] - cx;
  o[1] = X[idx * 3 + 1] - cy;
  o[2] = X[idx * 3 + 2] - cz;
  const float* q = pts + ((size_t)b * NPTS + idx) * 6;
#pragma unroll
  for (int c = 0; c < 6; ++c) o[3 + c] = q[c];
#pragma unroll
  for (int c = 9; c < 16; ++c) o[c] = 0.f;    // zero K-padding
}

// ---------------- WMMA fp32 GEMM for layers 0/1 (Cout=64) -------------------
// Wave = one 16-row tile x 64 cols; applies fused BN(a,c)+ReLU of the previous
// layer at A-load time; writes raw y; accumulates channel sum/sum^2 via LDS.
template <int KIN, int RK, bool ACT>
__global__ __launch_bounds__(256) void gemm_layer(const float* __restrict__ Xin,
                                                  const float* __restrict__ W,   // (64, RK)
                                                  const float2* __restrict__ ab,
                                                  float* __restrict__ Yout,
                                                  float* __restrict__ stats) {
  constexpr int COUT = 64;
  __shared__ float sS[COUT], sS2[COUT];
  const int tid = threadIdx.x;
  if (tid < COUT) { sS[tid] = 0.f; sS2[tid] = 0.f; }
  __syncthreads();
  const int w = tid >> 5, lane = tid & 31;
  const int rowBase = (blockIdx.x * 8 + w) * 16;
  const int lrow = lane & 15;
  const int khalf = (lane >> 4) * 2;          // A: lanes 16-31 hold K+2,K+3
  const int rhalf = (lane >> 4) * 8;          // C: lanes 16-31 hold rows v+8
  v8f acc[4];
#pragma unroll
  for (int n = 0; n < 4; ++n)
#pragma unroll
    for (int v = 0; v < 8; ++v) acc[n][v] = 0.f;
  const float* Xr = Xin + (size_t)(rowBase + lrow) * KIN;
#pragma unroll
  for (int kt = 0; kt < KIN / 4; ++kt) {
    const int k0 = kt * 4 + khalf;
    float x0 = Xr[k0], x1 = Xr[k0 + 1];
    if (ACT) {
      const float2 s0 = ab[k0], s1 = ab[k0 + 1];
      x0 = relu_f(s0.x * x0 + s0.y);
      x1 = relu_f(s1.x * x1 + s1.y);
    }
    v2f a; a.x = x0; a.y = x1;
#pragma unroll
    for (int n = 0; n < 4; ++n) {
      const int col = n * 16 + lrow;
      v2f bb;                                  // B[k][col] = W[col][k], zero-padded
      bb.x = (k0 < RK)     ? W[col * RK + k0]     : 0.f;
      bb.y = (k0 + 1 < RK) ? W[col * RK + k0 + 1] : 0.f;
      acc[n] = __builtin_amdgcn_wmma_f32_16x16x4_f32(false, a, false, bb,
                                                     (short)0, acc[n], false, false);
    }
  }
#pragma unroll
  for (int n = 0; n < 4; ++n) {
    const int col = n * 16 + lrow;
    float ls = 0.f, ls2 = 0.f;
#pragma unroll
    for (int v = 0; v < 8; ++v) {
      const float y = acc[n][v];
      Yout[(size_t)(rowBase + rhalf + v) * COUT + col] = y;
      ls += y; ls2 += y * y;
    }
    atomicAdd(&sS[col], ls);
    atomicAdd(&sS2[col], ls2);
  }
  __syncthreads();
  if (tid < COUT) {
    atomicAdd(&stats[tid], sS[tid]);
    atomicAdd(&stats[COUT + tid], sS2[tid]);
  }
}

// ---------------- layer 2 GEMM fused with max/min-over-S epilogue -----------
// Wave = one (b,p) group (32 rows = two row tiles) x 64 cols (half of 128).
__global__ __launch_bounds__(256) void gemm_final(const float* __restrict__ Y1,
                                                  const float* __restrict__ W,   // (128,64)
                                                  const float2* __restrict__ ab,
                                                  float* __restrict__ gmax,
                                                  float* __restrict__ gmin,
                                                  float* __restrict__ stats) {
  constexpr int KIN = 64, C2 = 128;
  __shared__ float sS[C2], sS2[C2];
  const int tid = threadIdx.x;
  if (tid < C2) { sS[tid] = 0.f; sS2[tid] = 0.f; }
  __syncthreads();
  const int w = tid >> 5, lane = tid & 31;
  const int g = blockIdx.x * 8 + w;           // 0..16383
  const int bp = g >> 1;
  const int cg = g & 1;                       // column half
  const int lrow = lane & 15;
  const int khalf = (lane >> 4) * 2;
  v8f acc[2][4];
#pragma unroll
  for (int rt = 0; rt < 2; ++rt)
#pragma unroll
    for (int n = 0; n < 4; ++n)
#pragma unroll
      for (int v = 0; v < 8; ++v) acc[rt][n][v] = 0.f;
  const float* Xr0 = Y1 + (size_t)(bp * 32 + lrow) * KIN;        // samples 0..15
  const float* Xr1 = Y1 + (size_t)(bp * 32 + 16 + lrow) * KIN;   // samples 16..31
#pragma unroll
  for (int kt = 0; kt < KIN / 4; ++kt) {
    const int k0 = kt * 4 + khalf;
    const float2 s0 = ab[k0], s1 = ab[k0 + 1];
    v2f a0, a1;
    a0.x = relu_f(s0.x * Xr0[k0] + s0.y);
    a0.y = relu_f(s1.x * Xr0[k0 + 1] + s1.y);
    a1.x = relu_f(s0.x * Xr1[k0] + s0.y);
    a1.y = relu_f(s1.x * Xr1[k0 + 1] + s1.y);
#pragma unroll
    for (int n = 0; n < 4; ++n) {
      const int col = cg * 64 + n * 16 + lrow;
      v2f bb; bb.x = W[col * KIN + k0]; bb.y = W[col * KIN + k0 + 1];
      acc[0][n] = __builtin_amdgcn_wmma_f32_16x16x4_f32(false, a0, false, bb,
                                                        (short)0, acc[0][n], false, false);
      acc[1][n] = __builtin_amdgcn_wmma_f32_16x16x4_f32(false, a1, false, bb,
                                                        (short)0, acc[1][n], false, false);
    }
  }
#pragma unroll
  for (int n = 0; n < 4; ++n) {
    const int col = cg * 64 + n * 16 + lrow;
    float mx = -3.4e38f, mn = 3.4e38f, ls = 0.f, ls2 = 0.f;
#pragma unroll
    for (int rt = 0; rt < 2; ++rt)
#pragma unroll
      for (int v = 0; v < 8; ++v) {
        const float y = acc[rt][n][v];
        mx = fmaxf(mx, y); mn = fminf(mn, y);
        ls += y; ls2 += y * y;
      }
    mx = fmaxf(mx, __shfl_xor(mx, 16));       // combine complementary row halves
    mn = fminf(mn, __shfl_xor(mn, 16));
    atomicAdd(&sS[col], ls);
    atomicAdd(&sS2[col], ls2);
    if (lane < 16) {
      gmax[(size_t)bp * C2 + col] = mx;
      gmin[(size_t)bp * C2 + col] = mn;
    }
  }
  __syncthreads();
  if (tid < C2) {
    atomicAdd(&stats[tid], sS[tid]);
    atomicAdd(&stats[C2 + tid], sS2[tid]);
  }
}

// ---------------- fold BN stats into per-channel affine (a, c) --------------
__global__ void finalize_ab_kernel(const float* __restrict__ stats,
                                   const float* __restrict__ gamma,
                                   const float* __restrict__ beta,
                                   float2* __restrict__ ab, int C) {
  const int c = threadIdx.x;
  if (c >= C) return;
  const float inv_m = 1.f / (float)MROWS;
  const float mean = stats[c] * inv_m;
  const float var  = stats[C + c] * inv_m - mean * mean;
  const float a = gamma[c] * rsqrtf(var + BN_EPS);
  ab[c] = make_float2(a, beta[c] - mean * a);
}

// ---------------- final: relu(affine(max-or-min)) -> new_points -------------
__global__ __launch_bounds__(256) void writeout_kernel(const float* __restrict__ gmax,
                                                       const float* __restrict__ gmin,
                                                       const float2* __restrict__ ab,
                                                       float* __restrict__ out) {
  const int i = blockIdx.x * 256 + threadIdx.x;   // 0 .. BPC*128
  const int c = i & 127;
  const float2 s = ab[c];
  const float y = (s.x >= 0.f) ? gmax[i] : gmin[i];  // monotone BN affine
  out[i] = fmaxf(s.x * y + s.y, 0.f);
}

extern "C" void kernel_launch(void* const* d_in, const int* in_sizes, int n_in,
                              void* d_out, int out_size, void* d_ws, size_t ws_size,
                              hipStream_t stream) {
  (void)in_sizes; (void)n_in; (void)out_size; (void)ws_size;
  const float* xyz = (const float*)d_in[0];
  const float* pts = (const float*)d_in[1];
  const float* w0  = (const float*)d_in[2];
  const float* g0  = (const float*)d_in[4];
  const float* be0 = (const float*)d_in[5];
  const float* w1  = (const float*)d_in[6];
  const float* g1  = (const float*)d_in[8];
  const float* be1 = (const float*)d_in[9];
  const float* w2  = (const float*)d_in[10];
  const float* g2  = (const float*)d_in[12];
  const float* be2 = (const float*)d_in[13];

  float* out    = (float*)d_out;
  float* cent   = out;                       // (B, P, 3)
  float* newpts = out + (size_t)BPC * 3;     // (B, P, 128)

  float* ws   = (float*)d_ws;                // requires ~160 MB workspace
  float* A0   = ws + OFF_A0;
  float* Y0   = ws + OFF_Y0;
  float* Y1   = ws + OFF_Y1;
  float* gmax = ws + OFF_GMAX;
  float* gmin = ws + OFF_GMIN;
  float* st   = ws + OFF_STATS;
  float* st0 = st, * st1 = st + 256, * st2 = st + 512;
  float2* abb = (float2*)(ws + OFF_AB);
  float2* ab0 = abb, * ab1 = abb + 128, * ab2 = abb + 256;

  init_stats_kernel<<<3, 256, 0, stream>>>(st);
  fps_kernel<<<NB, 1024, 0, stream>>>(xyz, cent);
  group_kernel<<<BPC / 8, 256, 0, stream>>>(xyz, pts, cent, A0);
  gemm_layer<16, 9, false><<<MROWS / 128, 256, 0, stream>>>(A0, w0, nullptr, Y0, st0);
  finalize_ab_kernel<<<1, 128, 0, stream>>>(st0, g0, be0, ab0, 64);
  gemm_layer<64, 64, true><<<MROWS / 128, 256, 0, stream>>>(Y0, w1, ab0, Y1, st1);
  finalize_ab_kernel<<<1, 128, 0, stream>>>(st1, g1, be1, ab1, 64);
  gemm_final<<<(BPC * 2) / 8, 256, 0, stream>>>(Y1, w2, ab1, gmax, gmin, st2);
  finalize_ab_kernel<<<1, 128, 0, stream>>>(st2, g2, be2, ab2, 128);
  writeout_kernel<<<(BPC * 128) / 256, 256, 0, stream>>>(gmax, gmin, ab2, newpts);
}